// GAT_37151467110676
// MI455X (gfx1250) — compile-verified
//
#include <hip/hip_runtime.h>
#include <math.h>

#define TPB 256

typedef __attribute__((ext_vector_type(2))) float v2f;
typedef __attribute__((ext_vector_type(8))) float v8f;

__device__ __forceinline__ float elu_f(float x)   { return x > 0.f ? x : (expf(x) - 1.f); }
__device__ __forceinline__ float lrelu_f(float x) { return x > 0.f ? x : 0.2f * x; }

__device__ __forceinline__ void edge_sd(const int* __restrict__ ei, int E, int e,
                                        int& s, int& d) {
  if (e < E) { s = ei[e]; d = ei[E + e]; }
  else       { s = e - E; d = s; }          // self loops appended after real edges
}

// float atomic max via sign-split integer atomics (valid for all finite floats,
// destination initialized to -inf)
__device__ __forceinline__ void atomic_max_f32(float* addr, float v) {
  if (v >= 0.f) atomicMax((int*)addr, __float_as_int(v));
  else          atomicMin((unsigned int*)addr, __float_as_uint(v));
}

__device__ __forceinline__ void atomic_add_f32(float* addr, float v) {
  __hip_atomic_fetch_add(addr, v, __ATOMIC_RELAXED, __HIP_MEMORY_SCOPE_AGENT);
}

// ---------------------------------------------------------------- utility fill
__global__ void fill_f32(float* __restrict__ p, float v, long n) {
  long i = (long)blockIdx.x * blockDim.x + threadIdx.x;
  if (i < n) p[i] = v;
}

// ------------------------------------------------ layer 0 GEMM: h = x @ W0^T (K=2)
__global__ void gemm_k2(const float* __restrict__ X, const float* __restrict__ W,
                        float* __restrict__ Hh, int n, int mdim) {
  long i = (long)blockIdx.x * blockDim.x + threadIdx.x;
  if (i >= (long)n * mdim) return;
  int nn = (int)(i / mdim), mm = (int)(i % mdim);
  float x0 = X[(size_t)nn * 2 + 0], x1 = X[(size_t)nn * 2 + 1];
  Hh[i] = fmaf(x0, W[(size_t)mm * 2 + 0], x1 * W[(size_t)mm * 2 + 1]);
}

// --------------------------------------- layer 1 GEMM via V_WMMA_F32_16X16X4_F32
// h(n,512) = X(n,128) @ W(512,128)^T. One wave -> one 16x16 tile, 32 K-steps of 4.
// A-frag layout (32-bit 16x4): lanes 0-15 hold K=0,1; lanes 16-31 hold K=2,3.
// C/D layout: VGPR i -> row (i + 8*(lane>=16)), col = lane&15.
#define XS_STRIDE 132   // 132 floats = 528 B: kills stride-128 bank conflicts, 16B-divisible
__global__ void __launch_bounds__(256)
gemm_wmma_f32_k128(const float* __restrict__ X, const float* __restrict__ W,
                   float* __restrict__ Hh, int n) {
  __shared__ float xs[16 * XS_STRIDE];
  const int tid  = threadIdx.x;
  const int row0 = blockIdx.x * 16;
  const bool full_tile = (row0 + 16 <= n);   // block-uniform -> scalar branch

  if (full_tile) {
    // vectorized staging: 16 rows x 32 float4 = 512 float4, 2 per thread
    const float4* xg = (const float4*)(X + (size_t)row0 * 128);
    for (int i = tid; i < 16 * 32; i += 256) {
      int r = i >> 5, c4 = i & 31;
      *(float4*)(xs + r * XS_STRIDE + (c4 << 2)) = xg[(r << 5) + c4];
    }
  } else {
    for (int i = tid; i < 16 * 128; i += 256) {
      int r = i >> 7, c = i & 127;
      int gr = row0 + r;
      xs[r * XS_STRIDE + c] = (gr < n) ? X[(size_t)gr * 128 + c] : 0.f;
    }
  }
  __syncthreads();

  const int wave = tid >> 5;
  const int lane = tid & 31;
  const int l16  = lane & 15;
  const int kOff = (lane >> 4) << 1;        // 0 (lanes 0-15) or 2 (lanes 16-31)
  const int col0 = (blockIdx.y * 8 + wave) * 16;

  const float* arow = xs + l16 * XS_STRIDE;           // A row (node) for this lane
  const float* brow = W + (size_t)(col0 + l16) * 128; // B col = row of W (W^T col)

  v8f acc = {};
#pragma unroll 8
  for (int k = 0; k < 128; k += 4) {
    v2f a, b;
    a.x = arow[k + kOff];
    a.y = arow[k + kOff + 1];
    b.x = brow[k + kOff];
    b.y = brow[k + kOff + 1];
    acc = __builtin_amdgcn_wmma_f32_16x16x4_f32(false, a, false, b, (short)0, acc,
                                                false, false);
  }

  const int rbase = row0 + ((lane >> 4) << 3);
  float* out = Hh + (size_t)rbase * 512 + col0 + l16;
  if (full_tile) {
#pragma unroll
    for (int i = 0; i < 8; ++i) out[(size_t)i * 512] = acc[i];
  } else {
#pragma unroll
    for (int i = 0; i < 8; ++i)
      if (rbase + i < n) out[(size_t)i * 512] = acc[i];
  }
}

// ------------------------------------------- layer 2 GEMM: h2 = x @ W2^T (M=3)
__global__ void gemm_k128_small(const float* __restrict__ X, const float* __restrict__ W,
                                float* __restrict__ Hh, int n, int mdim) {
  long i = (long)blockIdx.x * blockDim.x + threadIdx.x;
  if (i >= (long)n * mdim) return;
  int nn = (int)(i / mdim), mm = (int)(i % mdim);
  const float* xr = X + (size_t)nn * 128;
  const float* wr = W + (size_t)mm * 128;
  float acc = 0.f;
#pragma unroll 8
  for (int k = 0; k < 128; ++k) acc = fmaf(xr[k], wr[k], acc);
  Hh[i] = acc;
}

// ---------------------------- per-(node,head) attention dots: ssum/dsum = <h, a>
__global__ void node_attn_dots(const float* __restrict__ Hh,
                               const float* __restrict__ as_, const float* __restrict__ ad_,
                               float* __restrict__ ssum, float* __restrict__ dsum,
                               int Nn, int Hheads, int C) {
  int idx = blockIdx.x * blockDim.x + threadIdx.x;
  if (idx >= Nn * Hheads) return;
  int n = idx / Hheads, hh = idx - n * Hheads;
  const float* hp = Hh + ((size_t)n * Hheads + hh) * C;
  const float* ap = as_ + (size_t)hh * C;
  const float* bp = ad_ + (size_t)hh * C;
  float s = 0.f, d = 0.f;
  for (int c = 0; c < C; ++c) { float v = hp[c]; s = fmaf(v, ap[c], s); d = fmaf(v, bp[c], d); }
  ssum[idx] = s; dsum[idx] = d;
}

// ---------------------------------------- edge pass 1: logits + segment max(dst)
__global__ void edge_logit_max(const int* __restrict__ ei, int E, int Nn, int Hheads,
                               const float* __restrict__ ssum, const float* __restrict__ dsum,
                               float* __restrict__ elog, float* __restrict__ mrow) {
  int idx = blockIdx.x * blockDim.x + threadIdx.x;
  int total = (E + Nn) * Hheads;
  if (idx >= total) return;
  int e = idx / Hheads, hh = idx - e * Hheads;
  int s, d; edge_sd(ei, E, e, s, d);
  float lg = lrelu_f(ssum[s * Hheads + hh] + dsum[d * Hheads + hh]);
  elog[idx] = lg;
  atomic_max_f32(&mrow[d * Hheads + hh], lg);
}

// ------------------------------------- edge pass 2: p = exp(l - m), segment sum
__global__ void edge_exp_sum(const int* __restrict__ ei, int E, int Nn, int Hheads,
                             const float* __restrict__ mrow, float* __restrict__ elog,
                             float* __restrict__ psum) {
  int idx = blockIdx.x * blockDim.x + threadIdx.x;
  int total = (E + Nn) * Hheads;
  if (idx >= total) return;
  int e = idx / Hheads, hh = idx - e * Hheads;
  int s, d; edge_sd(ei, E, e, s, d);
  float p = expf(elog[idx] - mrow[d * Hheads + hh]);
  elog[idx] = p;
  atomic_add_f32(&psum[d * Hheads + hh], p);
}

// --------------------------- edge pass 3: agg[dst] += h[src] * alpha (scatter)
// C % 4 == 0 path uses float4 gathers (rows are 512B aligned) -> b128 vmem loads.
__global__ void edge_aggregate(const int* __restrict__ ei, int E, int Nn, int Hheads, int C,
                               const float* __restrict__ Hh, const float* __restrict__ elog,
                               const float* __restrict__ psum, float* __restrict__ agg) {
  int idx = blockIdx.x * blockDim.x + threadIdx.x;
  int total = (E + Nn) * Hheads;
  if (idx >= total) return;
  int e = idx / Hheads, hh = idx - e * Hheads;
  int s, d; edge_sd(ei, E, e, s, d);
  float alpha = elog[idx] / (psum[d * Hheads + hh] + 1e-16f);
  const float* hs = Hh + ((size_t)s * Hheads + hh) * C;
  float* ag = agg + ((size_t)d * Hheads + hh) * C;
  if ((C & 3) == 0) {
    const float4* hv = (const float4*)hs;
    for (int c4 = 0; c4 < (C >> 2); ++c4) {
      float4 v = hv[c4];
      float* a4 = ag + (c4 << 2);
      atomic_add_f32(a4 + 0, v.x * alpha);
      atomic_add_f32(a4 + 1, v.y * alpha);
      atomic_add_f32(a4 + 2, v.z * alpha);
      atomic_add_f32(a4 + 3, v.w * alpha);
    }
  } else {
    for (int c = 0; c < C; ++c) atomic_add_f32(&ag[c], hs[c] * alpha);
  }
}

// ------------------- mid-layer epilogue: mean heads + bias -> ELU -> BatchNorm
__global__ void node_post_mid(const float* __restrict__ agg, const float* __restrict__ bias,
                              const float* __restrict__ g, const float* __restrict__ bb,
                              const float* __restrict__ mm, const float* __restrict__ vv,
                              float* __restrict__ xout, int Nn, int Hheads, int C) {
  long idx = (long)blockIdx.x * blockDim.x + threadIdx.x;
  if (idx >= (long)Nn * C) return;
  int n = (int)(idx / C), c = (int)(idx % C);
  float s = 0.f;
  for (int hh = 0; hh < Hheads; ++hh) s += agg[((size_t)n * Hheads + hh) * C + c];
  s = s / (float)Hheads + bias[c];
  s = elu_f(s);
  s = (s - mm[c]) * rsqrtf(vv[c] + 1e-5f) * g[c] + bb[c];
  xout[idx] = s;
}

// -------------------- layer-2 epilogue (concat, H=1) fused with dynamics model
__global__ void final_dynamics(const float* __restrict__ agg, const float* __restrict__ b2,
                               const float* __restrict__ k_, const float* __restrict__ d_,
                               const float* __restrict__ t0_, const float* __restrict__ u0_,
                               const float* __restrict__ t, float* __restrict__ out, int Nn) {
  int n = blockIdx.x * blockDim.x + threadIdx.x;
  if (n >= Nn) return;
  float a_r = elu_f(agg[(size_t)n * 3 + 0] + b2[0]);
  float gam = elu_f(agg[(size_t)n * 3 + 1] + b2[1]);
  float bet = elu_f(agg[(size_t)n * 3 + 2] + b2[2]);
  float tt = t[n];
  float K = k_[0], D = d_[0], T0 = t0_[0], U0 = u0_[0];
  float S   = 1.f / (1.f + expf(-(K * (tt - T0 - D))));
  float eb  = expf(-bet * tt),        eg  = expf(-gam * tt);
  float ebs = expf(-bet * (tt - T0)), egs = expf(-gam * (tt - T0));
  float ab = a_r / bet, ag = a_r / gam;
  float tu = ab * (1.f - eb) * (1.f - S) + ab * S + (U0 * ebs - ab) * S;
  float inv = a_r / (gam - bet);
  float ts = (ag * (1.f - eg) + inv * (eg - eb)) * (1.f - S) + ag * S
           + bet * U0 / (gam - bet) * (egs - ebs) * S;
  out[n] = tu;
  out[Nn + n] = ts;
}

static inline int cdiv(long a, long b) { return (int)((a + b - 1) / b); }

extern "C" void kernel_launch(void* const* d_in, const int* in_sizes, int n_in,
                              void* d_out, int out_size, void* d_ws, size_t ws_size,
                              hipStream_t stream) {
  const float* x    = (const float*)d_in[0];
  const int*   ei   = (const int*)d_in[1];
  /* d_in[2] = gene_index (always 0; all scalar params are size-1 arrays) */
  const float* W0   = (const float*)d_in[3];
  const float* as0  = (const float*)d_in[4];
  const float* ad0  = (const float*)d_in[5];
  const float* b0   = (const float*)d_in[6];
  const float* W1   = (const float*)d_in[7];
  const float* as1  = (const float*)d_in[8];
  const float* ad1  = (const float*)d_in[9];
  const float* b1   = (const float*)d_in[10];
  const float* W2   = (const float*)d_in[11];
  const float* as2  = (const float*)d_in[12];
  const float* ad2  = (const float*)d_in[13];
  const float* b2   = (const float*)d_in[14];
  const float* bng0 = (const float*)d_in[15];
  const float* bnb0 = (const float*)d_in[16];
  const float* bnm0 = (const float*)d_in[17];
  const float* bnv0 = (const float*)d_in[18];
  const float* bng1 = (const float*)d_in[19];
  const float* bnb1 = (const float*)d_in[20];
  const float* bnm1 = (const float*)d_in[21];
  const float* bnv1 = (const float*)d_in[22];
  const float* kk   = (const float*)d_in[23];
  const float* dd   = (const float*)d_in[24];
  const float* t0   = (const float*)d_in[25];
  const float* u0   = (const float*)d_in[26];
  const float* t    = (const float*)d_in[27];

  const int N = in_sizes[0] / 2;      // x is (N, 2)
  const int E = in_sizes[1] / 2;      // edge_index is (2, E)
  const int Etot = E + N;

  float* ws = (float*)d_ws;
  size_t off = 0;
  float* h    = ws + off; off += (size_t)N * 512;   // GEMM output (reused all layers)
  float* agg  = ws + off; off += (size_t)N * 512;   // scatter destination
  float* x1   = ws + off; off += (size_t)N * 128;   // node features between layers
  float* elog = ws + off; off += (size_t)Etot * 4;  // per-edge logits / probs
  float* mrow = ws + off; off += (size_t)N * 4;     // segment max
  float* psum = ws + off; off += (size_t)N * 4;     // segment sum
  float* ssum = ws + off; off += (size_t)N * 4;     // <h, a_s>
  float* dsum = ws + off; off += (size_t)N * 4;     // <h, a_d>

  // ================= Layer 0 (H=4, C=128, K=2) =================
  {
    const int H = 4, C = 128;
    gemm_k2<<<cdiv((long)N * 512, TPB), TPB, 0, stream>>>(x, W0, h, N, 512);
    node_attn_dots<<<cdiv((long)N * H, TPB), TPB, 0, stream>>>(h, as0, ad0, ssum, dsum, N, H, C);
    fill_f32<<<cdiv((long)N * H, TPB), TPB, 0, stream>>>(mrow, -INFINITY, (long)N * H);
    fill_f32<<<cdiv((long)N * H, TPB), TPB, 0, stream>>>(psum, 0.f, (long)N * H);
    fill_f32<<<cdiv((long)N * H * C, TPB), TPB, 0, stream>>>(agg, 0.f, (long)N * H * C);
    edge_logit_max<<<cdiv((long)Etot * H, TPB), TPB, 0, stream>>>(ei, E, N, H, ssum, dsum, elog, mrow);
    edge_exp_sum  <<<cdiv((long)Etot * H, TPB), TPB, 0, stream>>>(ei, E, N, H, mrow, elog, psum);
    edge_aggregate<<<cdiv((long)Etot * H, TPB), TPB, 0, stream>>>(ei, E, N, H, C, h, elog, psum, agg);
    node_post_mid<<<cdiv((long)N * C, TPB), TPB, 0, stream>>>(agg, b0, bng0, bnb0, bnm0, bnv0, x1, N, H, C);
  }

  // ================= Layer 1 (H=4, C=128, K=128) — WMMA GEMM =================
  {
    const int H = 4, C = 128;
    dim3 grid(cdiv(N, 16), 4);   // 8 waves/block x 4 -> 32 column tiles = 512 cols
    gemm_wmma_f32_k128<<<grid, 256, 0, stream>>>(x1, W1, h, N);
    node_attn_dots<<<cdiv((long)N * H, TPB), TPB, 0, stream>>>(h, as1, ad1, ssum, dsum, N, H, C);
    fill_f32<<<cdiv((long)N * H, TPB), TPB, 0, stream>>>(mrow, -INFINITY, (long)N * H);
    fill_f32<<<cdiv((long)N * H, TPB), TPB, 0, stream>>>(psum, 0.f, (long)N * H);
    fill_f32<<<cdiv((long)N * H * C, TPB), TPB, 0, stream>>>(agg, 0.f, (long)N * H * C);
    edge_logit_max<<<cdiv((long)Etot * H, TPB), TPB, 0, stream>>>(ei, E, N, H, ssum, dsum, elog, mrow);
    edge_exp_sum  <<<cdiv((long)Etot * H, TPB), TPB, 0, stream>>>(ei, E, N, H, mrow, elog, psum);
    edge_aggregate<<<cdiv((long)Etot * H, TPB), TPB, 0, stream>>>(ei, E, N, H, C, h, elog, psum, agg);
    node_post_mid<<<cdiv((long)N * C, TPB), TPB, 0, stream>>>(agg, b1, bng1, bnb1, bnm1, bnv1, x1, N, H, C);
  }

  // ================= Layer 2 (H=1, C=3, K=128) + dynamics =================
  {
    const int H = 1, C = 3;
    gemm_k128_small<<<cdiv((long)N * 3, TPB), TPB, 0, stream>>>(x1, W2, h, N, 3);
    node_attn_dots<<<cdiv((long)N * H, TPB), TPB, 0, stream>>>(h, as2, ad2, ssum, dsum, N, H, C);
    fill_f32<<<cdiv((long)N * H, TPB), TPB, 0, stream>>>(mrow, -INFINITY, (long)N * H);
    fill_f32<<<cdiv((long)N * H, TPB), TPB, 0, stream>>>(psum, 0.f, (long)N * H);
    fill_f32<<<cdiv((long)N * H * C, TPB), TPB, 0, stream>>>(agg, 0.f, (long)N * H * C);
    edge_logit_max<<<cdiv((long)Etot * H, TPB), TPB, 0, stream>>>(ei, E, N, H, ssum, dsum, elog, mrow);
    edge_exp_sum  <<<cdiv((long)Etot * H, TPB), TPB, 0, stream>>>(ei, E, N, H, mrow, elog, psum);
    edge_aggregate<<<cdiv((long)Etot * H, TPB), TPB, 0, stream>>>(ei, E, N, H, C, h, elog, psum, agg);
    final_dynamics<<<cdiv(N, TPB), TPB, 0, stream>>>(agg, b2, kk, dd, t0, u0, t, (float*)d_out, N);
  }
}